// MSTAGNN_63333587746842
// MI455X (gfx1250) — compile-verified
//
#include <hip/hip_runtime.h>
#include <hip/hip_bf16.h>

#define NNODES 20000
#define NEDGES 160000
#define NFEDGES 500000
#define IN_DIM 128
#define HC 64
#define HEADS 4
#define HEADC 16
#define DV 16
#define KHOP 3

typedef __attribute__((ext_vector_type(2))) float v2f;
typedef __attribute__((ext_vector_type(8))) float v8f;

__device__ __forceinline__ float siluf(float v) { return v / (1.0f + __expf(-v)); }
__device__ __forceinline__ float eluf(float v)  { return v > 0.0f ? v : (__expf(v) - 1.0f); }

// C[16x16] (+)= A[16xK] * B[Kx16], A row-major stride lda, B row-major stride ldb.
// Uses V_WMMA_F32_16X16X4_F32 (full fp32, matches reference precision).
// A layout: lanes 0-15 row=lane K={k,k+1}; lanes 16-31 row=lane-16 K={k+2,k+3}.
// B layout: vgpr0 lanes0-15 row k, lanes16-31 row k+2; vgpr1 rows k+1/k+3.
__device__ __forceinline__ v8f wmma_f32_tile(const float* A, int lda,
                                             const float* B, int ldb,
                                             int K, v8f acc) {
  const int lane = threadIdx.x & 31;
  const int half = lane >> 4;
  const int l16  = lane & 15;
  for (int k = 0; k < K; k += 4) {
    v2f a, b;
    a.x = A[l16 * lda + (k + 2 * half + 0)];
    a.y = A[l16 * lda + (k + 2 * half + 1)];
    b.x = B[(k + 2 * half + 0) * ldb + l16];
    b.y = B[(k + 2 * half + 1) * ldb + l16];
    acc = __builtin_amdgcn_wmma_f32_16x16x4_f32(false, a, false, b, (short)0, acc,
                                                false, false);
  }
  return acc;
}

__global__ void zero_f32(float* p, long n) {
  long i  = (long)blockIdx.x * blockDim.x + threadIdx.x;
  long st = (long)gridDim.x * blockDim.x;
  for (; i < n; i += st) p[i] = 0.0f;
}

__global__ void deg_count(const int* __restrict__ col, float* __restrict__ deg, int E) {
  int i = blockIdx.x * blockDim.x + threadIdx.x;
  if (i < E) atomicAdd(&deg[col[i]], 1.0f);
}

__global__ void deg_invert(const float* __restrict__ deg, float* __restrict__ dinv, int N) {
  int i = blockIdx.x * blockDim.x + threadIdx.x;
  if (i < N) { float d = deg[i]; dinv[i] = (d > 0.0f) ? (1.0f / d) : 0.0f; }
}

// One wave per 16-node tile: t_emb -> MLP -> h -> Q/K/V -> M outer product + hidden init.
__global__ void __launch_bounds__(32)
node_pipeline(const float* __restrict__ x, const int* __restrict__ tsteps,
              const float* __restrict__ Wi, const float* __restrict__ bi,
              const float* __restrict__ Wt1, const float* __restrict__ bt1,
              const float* __restrict__ Wt2, const float* __restrict__ bt2,
              const float* __restrict__ WQ, const float* __restrict__ bQ,
              const float* __restrict__ WK, const float* __restrict__ bK,
              const float* __restrict__ WV, const float* __restrict__ bV,
              const float* __restrict__ hopwise,
              float* __restrict__ Qg, float* __restrict__ Kg,
              float* __restrict__ Mg, float* __restrict__ hiddeng) {
  __shared__ float s_temb[16 * 64];   // raw sinusoid, later final t_emb
  __shared__ float s_a1[16 * 256];    // MLP hidden
  __shared__ float s_h[16 * 64];
  __shared__ float s_kf[16 * 64];
  __shared__ float s_v[16 * 64];

  const int lane = threadIdx.x;
  const int half = lane >> 4, l16 = lane & 15;
  const int nb = blockIdx.x * 16;

  // ---- sinusoidal embedding (dim 64: 32 sin + 32 cos), t scaled by 4000/128
  for (int idx = lane; idx < 16 * 64; idx += 32) {
    int m = idx >> 6, c = idx & 63;
    float t = (float)tsteps[nb + m] * (4000.0f / 128.0f);
    int j = c & 31;
    float arg = t * __expf(-0.29710775393471563f * (float)j);  // -ln(1e4)/31 * j
    s_temb[idx] = (c < 32) ? __sinf(arg) : __cosf(arg);
  }
  __syncthreads();

  // ---- a1 = silu(temb @ Wt1 + bt1)   [16 x 256]
  for (int ct = 0; ct < 16; ++ct) {
    v8f acc = {0.f, 0.f, 0.f, 0.f, 0.f, 0.f, 0.f, 0.f};
    acc = wmma_f32_tile(s_temb, 64, Wt1 + ct * 16, 256, 64, acc);
    int cb = ct * 16;
    for (int r = 0; r < 8; ++r) {
      int m = r + 8 * half;
      s_a1[m * 256 + cb + l16] = siluf(acc[r] + bt1[cb + l16]);
    }
  }
  __syncthreads();

  // ---- t_emb = a1 @ Wt2 + bt2  (overwrite s_temb)
  for (int ct = 0; ct < 4; ++ct) {
    v8f acc = {0.f, 0.f, 0.f, 0.f, 0.f, 0.f, 0.f, 0.f};
    acc = wmma_f32_tile(s_a1, 256, Wt2 + ct * 16, 64, 256, acc);
    for (int r = 0; r < 8; ++r) {
      int m = r + 8 * half;
      s_temb[m * 64 + ct * 16 + l16] = acc[r] + bt2[ct * 16 + l16];
    }
  }
  __syncthreads();

  // ---- h = relu(x @ Wi + bi + t_emb)
  for (int ct = 0; ct < 4; ++ct) {
    v8f acc = {0.f, 0.f, 0.f, 0.f, 0.f, 0.f, 0.f, 0.f};
    acc = wmma_f32_tile(x + (long)nb * IN_DIM, IN_DIM, Wi + ct * 16, 64, IN_DIM, acc);
    for (int r = 0; r < 8; ++r) {
      int m = r + 8 * half, c = ct * 16 + l16;
      float z = acc[r] + bi[c] + s_temb[m * 64 + c];
      s_h[m * 64 + c] = fmaxf(z, 0.0f);
    }
  }
  __syncthreads();

  // ---- Q = 1+elu(h@WQ+bQ), Kf = 1+elu(h@WK+bK), V = h@WV+bV
  for (int ct = 0; ct < 4; ++ct) {
    v8f aq = {0.f, 0.f, 0.f, 0.f, 0.f, 0.f, 0.f, 0.f};
    v8f ak = {0.f, 0.f, 0.f, 0.f, 0.f, 0.f, 0.f, 0.f};
    v8f av = {0.f, 0.f, 0.f, 0.f, 0.f, 0.f, 0.f, 0.f};
    aq = wmma_f32_tile(s_h, 64, WQ + ct * 16, 64, 64, aq);
    ak = wmma_f32_tile(s_h, 64, WK + ct * 16, 64, 64, ak);
    av = wmma_f32_tile(s_h, 64, WV + ct * 16, 64, 64, av);
    for (int r = 0; r < 8; ++r) {
      int m = r + 8 * half, c = ct * 16 + l16;
      float q  = 1.0f + eluf(aq[r] + bQ[c]);
      float kf = 1.0f + eluf(ak[r] + bK[c]);
      float vv = av[r] + bV[c];
      Qg[(long)(nb + m) * 64 + c] = q;
      Kg[(long)(nb + m) * 64 + c] = kf;
      s_kf[m * 64 + c] = kf;
      s_v[m * 64 + c]  = vv;
    }
  }
  __syncthreads();

  // ---- hidden = V * hopwise[0]; M[n,hd,i,j] = Kf[n,hd,i]*V[n,hd,j]
  float h0 = hopwise[0];
  for (int idx = lane; idx < 16 * 64; idx += 32) {
    int m = idx >> 6, c = idx & 63;
    hiddeng[(long)(nb + m) * 64 + c] = s_v[m * 64 + c] * h0;
  }
  for (int idx = lane; idx < 16 * 1024; idx += 32) {
    int m = idx >> 10, r = idx & 1023;
    int hd = r >> 8, i = (r >> 4) & 15, j = r & 15;
    Mg[(long)(nb + m) * 1024 + r] = s_kf[m * 64 + hd * 16 + i] * s_v[m * 64 + hd * 16 + j];
  }
}

// One wave per edge: Mout[col] += Min[row]*w, Kout[col] += Kin[row]*w (atomic scatter).
__global__ void propagate(const int* __restrict__ row, const int* __restrict__ col,
                          const float* __restrict__ dinv,
                          const float* __restrict__ Min, const float* __restrict__ Kin,
                          float* __restrict__ Mout, float* __restrict__ Kout, int E) {
  int gw = (int)((blockIdx.x * (long)blockDim.x + threadIdx.x) >> 5);
  int lane = threadIdx.x & 31;
  if (gw >= E) return;
  int r = row[gw], c = col[gw];
  float w = dinv[r];
  if (w == 0.0f) return;
  const float* ms = Min + (long)r * 1024;
  float* md = Mout + (long)c * 1024;
  for (int it = 0; it < 32; ++it) {
    int idx = it * 32 + lane;
    atomicAdd(&md[idx], ms[idx] * w);
  }
  const float* ks = Kin + (long)r * 64;
  float* kd = Kout + (long)c * 64;
  int i2 = lane * 2;
  atomicAdd(&kd[i2],     ks[i2] * w);
  atomicAdd(&kd[i2 + 1], ks[i2 + 1] * w);
}

// hidden += gamma * (Q.M) / (Q.Kf + eps), thread per (n, head, j)
__global__ void hop_update(const float* __restrict__ Qg, const float* __restrict__ Mg,
                           const float* __restrict__ Kg, float* __restrict__ hiddeng,
                           const float* __restrict__ hopwise,
                           const float* __restrict__ headwise, int hop, int N) {
  int tid = blockIdx.x * blockDim.x + threadIdx.x;
  if (tid >= N * 64) return;
  int n = tid >> 6, hd = (tid >> 4) & 3, j = tid & 15;
  float e0 = headwise[0 * KHOP + hop], e1 = headwise[1 * KHOP + hop];
  float e2 = headwise[2 * KHOP + hop], e3 = headwise[3 * KHOP + hop];
  float mx = fmaxf(fmaxf(e0, e1), fmaxf(e2, e3));
  float ssum = __expf(e0 - mx) + __expf(e1 - mx) + __expf(e2 - mx) + __expf(e3 - mx);
  float gamma = hopwise[hop + 1] * (__expf(headwise[hd * KHOP + hop] - mx) / ssum);
  const float* q  = Qg + (long)n * 64 + hd * 16;
  const float* m  = Mg + (long)n * 1024 + hd * 256;
  const float* kf = Kg + (long)n * 64 + hd * 16;
  float H = 0.0f, C = 1e-5f;
  for (int i = 0; i < 16; ++i) {
    float qi = q[i];
    H += qi * m[i * 16 + j];
    C += qi * kf[i];
  }
  hiddeng[tid] += gamma * (H / C);
}

// hid16 = hidden[N,64] @ Wo[64,16] + bo   (one wave per 16 nodes)
__global__ void __launch_bounds__(32)
out_proj(const float* __restrict__ hiddeng, const float* __restrict__ Wo,
         const float* __restrict__ bo, float* __restrict__ hid16) {
  const int lane = threadIdx.x;
  const int half = lane >> 4, l16 = lane & 15;
  const int nb = blockIdx.x * 16;
  v8f acc = {0.f, 0.f, 0.f, 0.f, 0.f, 0.f, 0.f, 0.f};
  acc = wmma_f32_tile(hiddeng + (long)nb * 64, 64, Wo, 16, 64, acc);
  for (int r = 0; r < 8; ++r) {
    int m = r + 8 * half;
    hid16[(long)(nb + m) * 16 + l16] = acc[r] + bo[l16];
  }
}

// Edge MLP: logits = silu(concat(h[src],h[dst]) @ Wf1 + bf1) @ Wf2 + bf2
// One wave per 16 edges; layer 1 is a 16x32x16 WMMA GEMM.
__global__ void __launch_bounds__(32)
edge_head(const int* __restrict__ src, const int* __restrict__ dst,
          const float* __restrict__ hid16,
          const float* __restrict__ Wf1, const float* __restrict__ bf1,
          const float* __restrict__ Wf2, const float* __restrict__ bf2,
          float* __restrict__ out) {
  __shared__ float s_he[16 * 32];
  const int lane = threadIdx.x;
  const int half = lane >> 4, l16 = lane & 15;
  const long eb = (long)blockIdx.x * 16;
  for (int idx = lane; idx < 16 * 32; idx += 32) {
    int m = idx >> 5, c = idx & 31;
    int node = (c < 16) ? src[eb + m] : dst[eb + m];
    s_he[idx] = hid16[(long)node * 16 + (c & 15)];
  }
  __syncthreads();
  v8f acc = {0.f, 0.f, 0.f, 0.f, 0.f, 0.f, 0.f, 0.f};
  acc = wmma_f32_tile(s_he, 32, Wf1, 16, 32, acc);
  float w2 = Wf2[l16];
  float b2 = bf2[0];
  for (int r = 0; r < 8; ++r) {
    float val = siluf(acc[r] + bf1[l16]) * w2;
    for (int off = 1; off < 16; off <<= 1) val += __shfl_xor(val, off, 32);
    if (l16 == 0) out[eb + r + 8 * half] = val + b2;
  }
}

extern "C" void kernel_launch(void* const* d_in, const int* in_sizes, int n_in,
                              void* d_out, int out_size, void* d_ws, size_t ws_size,
                              hipStream_t stream) {
  const float* x    = (const float*)d_in[0];
  const int* eidx   = (const int*)d_in[1];   // [2,E]
  const int* feidx  = (const int*)d_in[2];   // [2,FE]
  const int* tsteps = (const int*)d_in[3];
  const float* Wi  = (const float*)d_in[4];
  const float* bi  = (const float*)d_in[5];
  const float* Wt1 = (const float*)d_in[6];
  const float* bt1 = (const float*)d_in[7];
  const float* Wt2 = (const float*)d_in[8];
  const float* bt2 = (const float*)d_in[9];
  const float* WQ  = (const float*)d_in[10];
  const float* bQ  = (const float*)d_in[11];
  const float* WK  = (const float*)d_in[12];
  const float* bK  = (const float*)d_in[13];
  const float* WV  = (const float*)d_in[14];
  const float* bV  = (const float*)d_in[15];
  const float* Wo  = (const float*)d_in[16];
  const float* bo  = (const float*)d_in[17];
  const float* hopwise  = (const float*)d_in[18];
  const float* headwise = (const float*)d_in[19];
  const float* Wf1 = (const float*)d_in[20];
  const float* bf1 = (const float*)d_in[21];
  const float* Wf2 = (const float*)d_in[22];
  const float* bf2 = (const float*)d_in[23];

  const int N = NNODES, E = NEDGES, FE = NFEDGES;
  const int* row = eidx;
  const int* col = eidx + E;

  float* ws = (float*)d_ws;
  float* deg  = ws;  ws += N;
  float* dinv = ws;  ws += N;
  float* Qg   = ws;  ws += (long)N * 64;
  float* K0   = ws;  ws += (long)N * 64;
  float* K1   = ws;  ws += (long)N * 64;
  float* hid  = ws;  ws += (long)N * 64;
  float* M0   = ws;  ws += (long)N * 1024;
  float* M1   = ws;  ws += (long)N * 1024;

  float* out_logits = (float*)d_out;       // [FE]
  float* out_hidden = out_logits + FE;     // [N,16]

  // degree / norm
  zero_f32<<<256, 256, 0, stream>>>(deg, N);
  deg_count<<<(E + 255) / 256, 256, 0, stream>>>(col, deg, E);
  deg_invert<<<(N + 255) / 256, 256, 0, stream>>>(deg, dinv, N);

  // fused node-feature pipeline (WMMA)
  node_pipeline<<<N / 16, 32, 0, stream>>>(x, tsteps, Wi, bi, Wt1, bt1, Wt2, bt2,
                                           WQ, bQ, WK, bK, WV, bV, hopwise,
                                           Qg, K0, M0, hid);

  float* Min = M0; float* Mout = M1;
  float* Kin = K0; float* Kout = K1;
  for (int hop = 0; hop < KHOP; ++hop) {
    zero_f32<<<4096, 256, 0, stream>>>(Mout, (long)N * 1024);
    zero_f32<<<512, 256, 0, stream>>>(Kout, (long)N * 64);
    propagate<<<(E + 7) / 8, 256, 0, stream>>>(row, col, dinv, Min, Kin, Mout, Kout, E);
    hop_update<<<(N * 64 + 255) / 256, 256, 0, stream>>>(Qg, Mout, Kout, hid,
                                                         hopwise, headwise, hop, N);
    float* t;
    t = Min; Min = Mout; Mout = t;
    t = Kin; Kin = Kout; Kout = t;
  }

  // output projection (WMMA) then edge regression head (WMMA layer 1)
  out_proj<<<N / 16, 32, 0, stream>>>(hid, Wo, bo, out_hidden);
  edge_head<<<FE / 16, 32, 0, stream>>>(feidx, feidx + FE, out_hidden,
                                        Wf1, bf1, Wf2, bf2, out_logits);
}